// CriterionEdgeKD_57878979280958
// MI455X (gfx1250) — compile-verified
//
#include <hip/hip_runtime.h>
#include <hip/hip_bf16.h>
#include <math.h>

// Problem sizes fixed by reference setup_inputs(): n=8, c=14, h=w=512
constexpr int N_  = 8;
constexpr int C_  = 14;
constexpr int H_  = 512;
constexpr int W_  = 512;
constexpr int HW_ = H_ * W_;          // 262144
constexpr int NHW_ = N_ * HW_;        // 2097152
constexpr int PAIRS_ = N_ * C_;       // 112 (n,c) pairs
constexpr int BLKS_  = 32;            // blocks per pair in the fused pass
constexpr int SLAB_  = HW_ / BLKS_;   // 8192 contiguous elems per block
constexpr int CHUNK_ = 4096;          // f32 elems per LDS-staged chunk (64KB/block)
constexpr int CHUNKS_ = SLAB_ / CHUNK_; // 2
constexpr float T1_ = 10.0f, T2_ = 50.0f;

typedef float v2f __attribute__((ext_vector_type(2)));
typedef float v8f __attribute__((ext_vector_type(8)));

#define DEV __device__ __forceinline__

// ---- 32-lane wave sum via CDNA5 V_WMMA_F32_16X16X4_F32 -------------------
// A (16x4 f32): VGPR0 lanes0-15 = {M=lane,K=0}, lanes16-31 = {M=lane-16,K=2};
// VGPR1 (K=1/K=3) zeroed. B = all-ones. D[m][*] = v[m] + v[m+16]; summing the
// 8 D VGPRs gives half-sums; shfl_xor(16) completes the 32-lane total.
DEV float waveSumWMMA(float v) {
    v2f a; a[0] = v;    a[1] = 0.0f;
    v2f b; b[0] = 1.0f; b[1] = 1.0f;
    v8f c = {};
    c = __builtin_amdgcn_wmma_f32_16x16x4_f32(false, a, false, b, (short)0, c, false, false);
    float h = ((c[0] + c[1]) + (c[2] + c[3])) + ((c[4] + c[5]) + (c[6] + c[7]));
    h += __shfl_xor(h, 16, 32);
    return h;
}

DEV float waveMax(float v) {
#pragma unroll
    for (int off = 16; off > 0; off >>= 1) v = fmaxf(v, __shfl_xor(v, off, 32));
    return v;
}

DEV int r101(int m, int L) {               // reflect-101 border (pad <= 5)
    if (m < 0) m = -m;
    if (m >= L) m = 2 * L - 2 - m;
    return m;
}

// ---------------- Tensor Data Mover (gfx1250) -----------------------------
#if defined(__HIP_DEVICE_COMPILE__) && __has_builtin(__builtin_amdgcn_tensor_load_to_lds)
#define USE_TDM 1
#else
#define USE_TDM 0
#endif

#if USE_TDM
typedef unsigned int v4u __attribute__((ext_vector_type(4)));
typedef int v4i __attribute__((ext_vector_type(4)));
typedef int v8i __attribute__((ext_vector_type(8)));

DEV unsigned ldsOffset(const void* p) {    // generic -> LDS byte offset
    return (unsigned)(unsigned long long)(const __attribute__((address_space(3))) void*)p;
}

// 1-D tile of `nelem` f32 from global -> LDS, per CDNA5 ISA D# layout (§8).
DEV void tdmLoad1D(const float* gsrc, unsigned lds_byte, int nelem) {
    unsigned long long ga = (unsigned long long)(size_t)gsrc;
    v4u g0;
    g0[0] = 1u;                                            // count=1 valid user D#
    g0[1] = lds_byte;                                      // lds_addr
    g0[2] = (unsigned)(ga & 0xffffffffu);                  // global_addr[31:0]
    g0[3] = (unsigned)((ga >> 32) & 0x01ffffffu) | (2u << 30); // addr[56:32] | type=2
    unsigned ne = (unsigned)nelem;
    v8i g1;
    g1[0] = (int)(2u << 16);                               // wg_mask=0, data_size=4B
    g1[1] = (int)((ne & 0xffffu) << 16);                   // tensor_dim0[15:0]
    g1[2] = (int)((ne >> 16) | (1u << 16));                // tensor_dim0[31:16] | tensor_dim1=1
    g1[3] = (int)((ne & 0xffffu) << 16);                   // tile_dim0 = nelem
    g1[4] = 0;                                             // tile_dim1=0 (1-D), tile_dim2=0
    g1[5] = (int)ne;                                       // tensor_dim0_stride[31:0]
    g1[6] = 0;
    g1[7] = 0;
    v4i z4 = {0, 0, 0, 0};
#if __has_include(<hip/amd_detail/amd_gfx1250_TDM.h>)
    v8i z8 = {0, 0, 0, 0, 0, 0, 0, 0};                     // therock 6-arg form
    __builtin_amdgcn_tensor_load_to_lds(g0, g1, z4, z4, z8, 0);
#else
    __builtin_amdgcn_tensor_load_to_lds(g0, g1, z4, z4, 0); // ROCm 7.2 5-arg form
#endif
}
#endif

// ---------------- 1) teacher argmax -> uint8-valued gray map --------------
__global__ __launch_bounds__(256) void k_argmax(const float* __restrict__ T, float* __restrict__ tmap) {
    int i = blockIdx.x * blockDim.x + threadIdx.x;
    if (i >= NHW_) return;
    int n = i / HW_, pix = i % HW_;
    const float* p = T + (size_t)n * C_ * HW_ + pix;
    float best = p[0]; int bi = 0;
#pragma unroll
    for (int c = 1; c < C_; ++c) {
        float v = p[(size_t)c * HW_];
        if (v > best) { best = v; bi = c; }        // first-max like jnp.argmax
    }
    tmap[i] = (float)((unsigned char)((float)bi / 13.0f * 255.0f)); // truncating cast
}

// ---------------- 2) Sobel (replicate border) + direction sector ----------
__global__ __launch_bounds__(256) void k_sobel(const float* __restrict__ tmap,
                                               float* __restrict__ mag,
                                               unsigned char* __restrict__ sec) {
    int i = blockIdx.x * blockDim.x + threadIdx.x;
    if (i >= NHW_) return;
    int n = i / HW_, p = i % HW_, y = p / W_, x = p % W_;
    const float* img = tmap + (size_t)n * HW_;
    auto AT = [&](int yy, int xx) -> float {
        yy = yy < 0 ? 0 : (yy >= H_ ? H_ - 1 : yy);
        xx = xx < 0 ? 0 : (xx >= W_ ? W_ - 1 : xx);
        return img[yy * W_ + xx];
    };
    float a00 = AT(y-1,x-1), a01 = AT(y-1,x), a02 = AT(y-1,x+1);
    float a10 = AT(y  ,x-1),                  a12 = AT(y  ,x+1);
    float a20 = AT(y+1,x-1), a21 = AT(y+1,x), a22 = AT(y+1,x+1);
    float gx = (a02 - a00) + 2.0f * (a12 - a10) + (a22 - a20);
    float gy = (a20 - a00) + 2.0f * (a21 - a01) + (a22 - a02);
    float m = fabsf(gx) + fabsf(gy);                       // L1 magnitude
    float ang = atan2f(gy, gx) * 57.29577951308232f;       // degrees
    ang = fmodf(ang, 180.0f); if (ang < 0.0f) ang += 180.0f;
    unsigned char s;
    if (ang < 22.5f || ang >= 157.5f) s = 0;
    else if (ang < 67.5f)             s = 1;
    else if (ang < 112.5f)            s = 2;
    else                              s = 3;
    mag[i] = m; sec[i] = s;
}

// ---------------- 3) non-max suppression + hysteresis thresholds ----------
__global__ __launch_bounds__(256) void k_nms(const float* __restrict__ mag,
                                             const unsigned char* __restrict__ sec,
                                             unsigned char* __restrict__ weak,
                                             unsigned char* __restrict__ strong) {
    int i = blockIdx.x * blockDim.x + threadIdx.x;
    if (i >= NHW_) return;
    int n = i / HW_, p = i % HW_, y = p / W_, x = p % W_;
    const float* mg = mag + (size_t)n * HW_;
    auto M = [&](int yy, int xx) -> float {                // zero-padded shifts
        if ((unsigned)yy >= (unsigned)H_ || (unsigned)xx >= (unsigned)W_) return 0.0f;
        return mg[yy * W_ + xx];
    };
    float m = mg[p], n1, n2;
    switch (sec[i]) {
        case 0:  n1 = M(y,   x+1); n2 = M(y,   x-1); break;
        case 1:  n1 = M(y-1, x+1); n2 = M(y+1, x-1); break;
        case 2:  n1 = M(y-1, x  ); n2 = M(y+1, x  ); break;
        default: n1 = M(y-1, x-1); n2 = M(y+1, x+1); break;
    }
    float nms = (m >= n1 && m >= n2) ? m : 0.0f;
    weak[i]   = nms > T1_ ? 1 : 0;
    strong[i] = nms > T2_ ? 1 : 0;
}

// ---------------- 4) hysteresis growth: strong = dilate3x3(strong) & weak --
__global__ __launch_bounds__(256) void k_hyst(const unsigned char* __restrict__ in,
                                              const unsigned char* __restrict__ weak,
                                              unsigned char* __restrict__ out) {
    int i = blockIdx.x * blockDim.x + threadIdx.x;
    if (i >= NHW_) return;
    int n = i / HW_, p = i % HW_, y = p / W_, x = p % W_;
    const unsigned char* img = in + (size_t)n * HW_;
    int g = 0;
#pragma unroll
    for (int dy = -1; dy <= 1; ++dy) {
        int yy = y + dy; if ((unsigned)yy >= (unsigned)H_) continue;
#pragma unroll
        for (int dx = -1; dx <= 1; ++dx) {
            int xx = x + dx; if ((unsigned)xx >= (unsigned)W_) continue;
            g |= img[yy * W_ + xx];
        }
    }
    out[i] = (g && weak[i]) ? 1 : 0;
}

// ---------------- 5) separable 10x10 dilate, window [-5,+4] ---------------
__global__ __launch_bounds__(256) void k_dilH(const unsigned char* __restrict__ in,
                                              unsigned char* __restrict__ out) {
    int i = blockIdx.x * blockDim.x + threadIdx.x;
    if (i >= NHW_) return;
    int n = i / HW_, p = i % HW_, y = p / W_, x = p % W_;
    const unsigned char* row = in + (size_t)n * HW_ + y * W_;
    int g = 0;
#pragma unroll
    for (int dx = -5; dx <= 4; ++dx) {
        int xx = x + dx; if ((unsigned)xx >= (unsigned)W_) continue;
        g |= row[xx];
    }
    out[i] = (unsigned char)g;
}
__global__ __launch_bounds__(256) void k_dilV(const unsigned char* __restrict__ in,
                                              float* __restrict__ out) {
    int i = blockIdx.x * blockDim.x + threadIdx.x;
    if (i >= NHW_) return;
    int n = i / HW_, p = i % HW_, y = p / W_, x = p % W_;
    const unsigned char* img = in + (size_t)n * HW_;
    int g = 0;
#pragma unroll
    for (int dy = -5; dy <= 4; ++dy) {
        int yy = y + dy; if ((unsigned)yy >= (unsigned)H_) continue;
        g |= img[yy * W_ + x];
    }
    out[i] = g ? 255.0f : 0.0f;
}

// ---------------- 6) separable 10x10 box blur, reflect-101 border ----------
__global__ __launch_bounds__(256) void k_blurH(const float* __restrict__ in, float* __restrict__ out) {
    int i = blockIdx.x * blockDim.x + threadIdx.x;
    if (i >= NHW_) return;
    int n = i / HW_, p = i % HW_, y = p / W_, x = p % W_;
    const float* row = in + (size_t)n * HW_ + y * W_;
    float s = 0.0f;
#pragma unroll
    for (int dx = -5; dx <= 4; ++dx) s += row[r101(x + dx, W_)];
    out[i] = s;
}
__global__ __launch_bounds__(256) void k_blurV(const float* __restrict__ in, float* __restrict__ edges) {
    int i = blockIdx.x * blockDim.x + threadIdx.x;
    if (i >= NHW_) return;
    int n = i / HW_, p = i % HW_, y = p / W_, x = p % W_;
    const float* img = in + (size_t)n * HW_;
    float s = 0.0f;
#pragma unroll
    for (int dy = -5; dy <= 4; ++dy) s += img[r101(y + dy, H_) * W_ + x];
    float b = s * 0.01f;                       // /100
    edges[i] = rintf(b) * (1.0f / 255.0f);     // jnp.round (RNE) then /255
}

// ---------------- 7) fused online-softmax KL pass -------------------------
// One HBM sweep over S/T. TDM double-buffers 4096-elem tiles of T and S into
// LDS (wave 0 drives DMA; s_wait_tensorcnt + barrier hands tiles to all 8
// waves). Per chunk: pass A masks in-place in LDS and finds chunk maxes (no
// exp); one rescale of running (z,sn) per chunk; pass B accumulates with
// native-rate __expf (args <= 0 by construction, so no overflow path needed).
// Wave combine: shfl max + WMMA ones-matmul sums; deterministic merges after.
__global__ __launch_bounds__(256) void k_fused(const float* __restrict__ S, const float* __restrict__ T,
                                               const float* __restrict__ edges,
                                               float* __restrict__ partials) {
    __shared__ float ldsT[2][CHUNK_];
    __shared__ float ldsS[2][CHUNK_];
    __shared__ float shm[8][5];

    const int pair = blockIdx.y;
    const int n = pair / C_;
    const int base = blockIdx.x * SLAB_;
    const float* pT = T + (size_t)pair * HW_ + base;
    const float* pS = S + (size_t)pair * HW_ + base;
    const float* e  = edges + (size_t)n * HW_ + base;
    const int tid = threadIdx.x;

    float mT = -INFINITY, zT = 0.0f, sn = 0.0f;
    float mS = -INFINITY, zS = 0.0f;

#if USE_TDM
    const bool issuer = (tid < 32);            // wave 0 drives the TDM
    if (issuer) {
        tdmLoad1D(pT, ldsOffset(&ldsT[0][0]), CHUNK_);
        tdmLoad1D(pS, ldsOffset(&ldsS[0][0]), CHUNK_);
    }
#endif
    for (int k = 0; k < CHUNKS_; ++k) {
        const int buf = k & 1;
#if USE_TDM
        if (issuer) {
            if (k + 1 < CHUNKS_) {             // prefetch next chunk, then wait
                tdmLoad1D(pT + (k + 1) * CHUNK_, ldsOffset(&ldsT[buf ^ 1][0]), CHUNK_);
                tdmLoad1D(pS + (k + 1) * CHUNK_, ldsOffset(&ldsS[buf ^ 1][0]), CHUNK_);
                __builtin_amdgcn_s_wait_tensorcnt(2);  // in-order: chunk k done
            } else {
                __builtin_amdgcn_s_wait_tensorcnt(0);
            }
        }
        __syncthreads();                       // chunk k visible to all waves
#else
        for (int j = tid; j < CHUNK_; j += 256) {
            ldsT[buf][j] = pT[k * CHUNK_ + j];
            ldsS[buf][j] = pS[k * CHUNK_ + j];
        }
        __syncthreads();
#endif
        __builtin_prefetch(e + (k + 1) * CHUNK_ + tid, 0, 0);  // -> global_prefetch_b8

        // pass A: mask in-place (own slots only -> no barrier needed), chunk maxes
        float cmT = -INFINITY, cmS = -INFINITY;
#pragma unroll
        for (int j = tid; j < CHUNK_; j += 256) {
            float ee = e[k * CHUNK_ + j];      // 8MB map: L2-resident across 14 c's
            float xT = ee * ldsT[buf][j];
            float xS = ee * ldsS[buf][j];
            ldsT[buf][j] = xT;
            ldsS[buf][j] = xS;
            cmT = fmaxf(cmT, xT);
            cmS = fmaxf(cmS, xS);
        }
        // one rescale per chunk (not per element)
        float nmT = fmaxf(mT, cmT);
        float rT  = __expf(mT - nmT);          // exp(-inf)=0 handles first chunk
        zT *= rT; sn *= rT; mT = nmT;
        float nmS = fmaxf(mS, cmS);
        zS *= __expf(mS - nmS); mS = nmS;

        // pass B: fixed-baseline accumulation, native-rate exp
#pragma unroll
        for (int j = tid; j < CHUNK_; j += 256) {
            float xT = ldsT[buf][j];
            float xS = ldsS[buf][j];
            float eT = __expf(xT - mT);
            zT += eT;
            sn += eT * (xT - xS);
            zS += __expf(xS - mS);
        }
        __syncthreads();                       // compute done before buf reuse
    }

    // ---- wave combine: shfl max, rescale, WMMA sums (EXEC all ones here) ----
    float mTw = waveMax(mT);
    float sT  = expf(mT - mTw);
    float zTw = waveSumWMMA(zT * sT);
    float snw = waveSumWMMA(sn * sT);
    float mSw = waveMax(mS);
    float zSw = waveSumWMMA(zS * expf(mS - mSw));

    const int wave = tid >> 5, lane = tid & 31;
    if (lane == 0) {
        shm[wave][0] = mTw; shm[wave][1] = zTw; shm[wave][2] = snw;
        shm[wave][3] = mSw; shm[wave][4] = zSw;
    }
    __syncthreads();
    if (tid == 0) {                            // deterministic 8-wave merge
        float MT = -INFINITY, MS = -INFINITY;
        for (int w = 0; w < 8; ++w) { MT = fmaxf(MT, shm[w][0]); MS = fmaxf(MS, shm[w][3]); }
        float ZT = 0.0f, SN = 0.0f, ZS = 0.0f;
        for (int w = 0; w < 8; ++w) {
            float r = expf(shm[w][0] - MT);
            ZT += shm[w][1] * r;
            SN += shm[w][2] * r;
            ZS += shm[w][4] * expf(shm[w][3] - MS);
        }
        float* o = partials + ((size_t)pair * BLKS_ + blockIdx.x) * 5;
        o[0] = MT; o[1] = ZT; o[2] = SN; o[3] = MS; o[4] = ZS;
    }
}

// ---------------- 8) finalize: merge block tuples -> scalar loss ----------
__global__ void k_finalize(const float* __restrict__ partials, float* __restrict__ out) {
    __shared__ float sh[128];
    int t = threadIdx.x;
    float contrib = 0.0f;
    if (t < PAIRS_) {
        const float* p = partials + (size_t)t * BLKS_ * 5;
        float MT = -INFINITY, MS = -INFINITY;
        for (int b = 0; b < BLKS_; ++b) { MT = fmaxf(MT, p[b*5+0]); MS = fmaxf(MS, p[b*5+3]); }
        float ZT = 0.0f, SN = 0.0f, ZS = 0.0f;
        for (int b = 0; b < BLKS_; ++b) {
            float r = expf(p[b*5+0] - MT);
            ZT += p[b*5+1] * r;
            SN += p[b*5+2] * r;
            ZS += p[b*5+4] * expf(p[b*5+3] - MS);
        }
        // sum_i p_t*(logsoftmax_t - logsoftmax_s)
        contrib = SN / ZT - MT - logf(ZT) + MS + logf(ZS);
    }
    sh[t] = contrib; __syncthreads();
    for (int s = 64; s > 0; s >>= 1) { if (t < s) sh[t] += sh[t + s]; __syncthreads(); }
    if (t == 0) out[0] = sh[0] * (1.0f / (float)PAIRS_);   // * TEMPERATURE^2 (=1)
}

extern "C" void kernel_launch(void* const* d_in, const int* in_sizes, int n_in,
                              void* d_out, int out_size, void* d_ws, size_t ws_size,
                              hipStream_t stream) {
    const float* S = (const float*)d_in[0];   // preds_S
    const float* T = (const float*)d_in[1];   // preds_T
    float* out = (float*)d_out;

    // workspace carve-up (~34 MB)
    float* tmap = (float*)d_ws;               // argmax map; reused as final edges
    float* mag  = tmap + NHW_;                // gradient mag; reused as blurH out
    float* dilf = mag  + NHW_;                // dilated 0/255 f32
    unsigned char* sec  = (unsigned char*)(dilf + NHW_);
    unsigned char* weak = sec  + NHW_;
    unsigned char* sA   = weak + NHW_;
    unsigned char* sB   = sA   + NHW_;
    float* partials = (float*)(sB + NHW_);    // PAIRS_*BLKS_*5 floats

    const int EB = 256;
    const int EG = (NHW_ + EB - 1) / EB;      // 8192

    k_argmax<<<EG, EB, 0, stream>>>(T, tmap);
    k_sobel <<<EG, EB, 0, stream>>>(tmap, mag, sec);
    k_nms   <<<EG, EB, 0, stream>>>(mag, sec, weak, sA);

    for (int it = 0; it < 16; ++it) {         // ends with result in sA (even count)
        const unsigned char* in = (it & 1) ? sB : sA;
        unsigned char* o        = (it & 1) ? sA : sB;
        k_hyst<<<EG, EB, 0, stream>>>(in, weak, o);
    }

    k_dilH <<<EG, EB, 0, stream>>>(sA, sB);   // sB free after hysteresis
    k_dilV <<<EG, EB, 0, stream>>>(sB, dilf);
    k_blurH<<<EG, EB, 0, stream>>>(dilf, mag);
    k_blurV<<<EG, EB, 0, stream>>>(mag, tmap);  // tmap now holds edges in [0,1]

    dim3 rg(BLKS_, PAIRS_);
    k_fused<<<rg, 256, 0, stream>>>(S, T, tmap, partials);
    k_finalize<<<1, 128, 0, stream>>>(partials, out);
}